// GCN_38560216384097
// MI455X (gfx1250) — compile-verified
//
#include <hip/hip_runtime.h>
#include <stdint.h>

// ---------------------------------------------------------------------------
// 2-layer GraphConv on MI455X (gfx1250).
//   agg1 = segment_sum(x[src] -> dst)        (hardware global_atomic_add_f32,
//                                             tables L2-resident: 192MB L2)
//   h    = relu(agg1 @ Wrel1^T + x @ Wroot1^T + b1)    (fp32 WMMA 16x16x4)
//   y2   = h @ Wrel2^T ; out = h @ Wroot2^T + b2       (fp32 WMMA)
//   out += segment_sum(y2[src] -> dst)       (64-dim: linear op commuted with
//                                             segment-sum to halve traffic)
// ---------------------------------------------------------------------------

#define N_NODES 100000
#define N_EDGES 1600000
#define D_IN    128
#define D_H     128
#define D_OUT   64

typedef float v2f __attribute__((ext_vector_type(2)));
typedef float v8f __attribute__((ext_vector_type(8)));

// Guaranteed CDNA5 hardware fp atomic (no-return form -> STOREcnt, RMW in L2
// atomic units). Avoids any chance of a compiler CAS-loop fallback.
__device__ __forceinline__ void atomic_add_f32_hw(float* p, float v) {
    asm volatile("global_atomic_add_f32 %0, %1, off"
                 :: "v"(p), "v"(v) : "memory");
}

// ---------------------------------------------------------------- zero fill
__global__ void gcn_zero_f32(float* __restrict__ p, int n) {
    int i = blockIdx.x * blockDim.x + threadIdx.x;
    if (i < n) p[i] = 0.0f;
}

// ------------------------------------------- scatter-add, 128 feats / edge
// one wave per edge; lane covers 4 consecutive features (b128 gather).
// Edge indices are a one-shot stream -> nontemporal so they don't evict the
// L2-resident node tables.
__global__ void gcn_scatter128(const float* __restrict__ x,
                               const long long* __restrict__ src,
                               const long long* __restrict__ dst,
                               float* __restrict__ agg) {
    long long tid = (long long)blockIdx.x * blockDim.x + threadIdx.x;
    int e    = (int)(tid >> 5);
    int lane = (int)(tid & 31);
    if (e >= N_EDGES) return;
    int s = (int)__builtin_nontemporal_load(&src[e]);
    int d = (int)__builtin_nontemporal_load(&dst[e]);
    const float4 v = ((const float4*)(x + (long long)s * D_IN))[lane];
    float* a = agg + (long long)d * D_IN + lane * 4;
    atomic_add_f32_hw(a + 0, v.x);
    atomic_add_f32_hw(a + 1, v.y);
    atomic_add_f32_hw(a + 2, v.z);
    atomic_add_f32_hw(a + 3, v.w);
}

// -------------------------------------------- scatter-add, 64 feats / edge
__global__ void gcn_scatter64(const float* __restrict__ y,
                              const long long* __restrict__ src,
                              const long long* __restrict__ dst,
                              float* __restrict__ out) {
    long long tid = (long long)blockIdx.x * blockDim.x + threadIdx.x;
    int e    = (int)(tid >> 5);
    int lane = (int)(tid & 31);
    if (e >= N_EDGES) return;
    int s = (int)__builtin_nontemporal_load(&src[e]);
    int d = (int)__builtin_nontemporal_load(&dst[e]);
    const float2 v = ((const float2*)(y + (long long)s * D_OUT))[lane];
    float* o = out + (long long)d * D_OUT + lane * 2;
    atomic_add_f32_hw(o + 0, v.x);
    atomic_add_f32_hw(o + 1, v.y);
}

// ----------------------------------------------------------- layer-1 GEMM
// block = 256 threads = 8 waves; block handles 16 node rows (N % 16 == 0),
// wave w computes the 16x16 output tile at columns [16w, 16w+16).
// D = agg(16xK) @ Wrel^T + x(16xK) @ Wroot^T  via v_wmma_f32_16x16x4_f32.
__global__ __launch_bounds__(256) void gcn_gemm1(
        const float* __restrict__ agg,    // [N,128]
        const float* __restrict__ x,      // [N,128]
        const float* __restrict__ Wrel,   // [128,128] row-major [out,in]
        const float* __restrict__ Wroot,  // [128,128]
        const float* __restrict__ bias,   // [128]
        float* __restrict__ h)            // [N,128]
{
    const int wave = threadIdx.x >> 5;       // 0..7 -> column tile
    const int lane = threadIdx.x & 31;
    const int half = lane >> 4;              // lane group (0/1)
    const int l16  = lane & 15;
    const int rowBase = blockIdx.x * 16;
    const int nBase   = wave * 16;

    // A frag: A[l16][k0 + 2*half + {0,1}]   (contiguous float2)
    const float* aRow  = agg   + (long long)(rowBase + l16) * D_IN + 2 * half;
    const float* xRow  = x     + (long long)(rowBase + l16) * D_IN + 2 * half;
    // B frag: W[n0 + l16][k0 + 2*half + {0,1}]
    const float* wRel  = Wrel  + (long long)(nBase + l16) * D_IN + 2 * half;
    const float* wRoot = Wroot + (long long)(nBase + l16) * D_IN + 2 * half;

    v8f acc = {};
#pragma unroll
    for (int k = 0; k < D_IN; k += 4) {
        v2f a = *(const v2f*)(aRow + k);
        v2f b = *(const v2f*)(wRel + k);
        acc = __builtin_amdgcn_wmma_f32_16x16x4_f32(false, a, false, b,
                                                    (short)0, acc, false, false);
    }
#pragma unroll
    for (int k = 0; k < D_IN; k += 4) {
        v2f a = *(const v2f*)(xRow + k);
        v2f b = *(const v2f*)(wRoot + k);
        acc = __builtin_amdgcn_wmma_f32_16x16x4_f32(false, a, false, b,
                                                    (short)0, acc, false, false);
    }

    const float bv = bias[nBase + l16];
#pragma unroll
    for (int v = 0; v < 8; ++v) {            // C/D: M = v + 8*half, N = l16
        const int m = v + 8 * half;
        const float r = acc[v] + bv;
        h[(long long)(rowBase + m) * D_H + nBase + l16] = fmaxf(r, 0.0f);
    }
}

// ----------------------------------------------------------- layer-2 GEMM
// waves 0..3: y2 = h @ Wrel2^T (64 cols); waves 4..7: out = h @ Wroot2^T + b2.
__global__ __launch_bounds__(256) void gcn_gemm2(
        const float* __restrict__ h,       // [N,128]
        const float* __restrict__ Wrel2,   // [64,128]
        const float* __restrict__ Wroot2,  // [64,128]
        const float* __restrict__ bias2,   // [64]
        float* __restrict__ y2,            // [N,64]
        float* __restrict__ out)           // [N,64] (d_out)
{
    const int wave = threadIdx.x >> 5;
    const int lane = threadIdx.x & 31;
    const int half = lane >> 4;
    const int l16  = lane & 15;
    const int rowBase = blockIdx.x * 16;
    const bool isRel  = wave < 4;
    const int nBase   = (wave & 3) * 16;

    const float* W    = isRel ? Wrel2 : Wroot2;
    const float* hRow = h + (long long)(rowBase + l16) * D_H + 2 * half;
    const float* wB   = W + (long long)(nBase + l16) * D_H + 2 * half;

    v8f acc = {};
#pragma unroll
    for (int k = 0; k < D_H; k += 4) {
        v2f a = *(const v2f*)(hRow + k);
        v2f b = *(const v2f*)(wB + k);
        acc = __builtin_amdgcn_wmma_f32_16x16x4_f32(false, a, false, b,
                                                    (short)0, acc, false, false);
    }

    float* dstp = isRel ? y2 : out;
    const float bv = isRel ? 0.0f : bias2[nBase + l16];
#pragma unroll
    for (int v = 0; v < 8; ++v) {
        const int m = v + 8 * half;
        dstp[(long long)(rowBase + m) * D_OUT + nBase + l16] = acc[v] + bv;
    }
}

// ---------------------------------------------------------------------------
extern "C" void kernel_launch(void* const* d_in, const int* in_sizes, int n_in,
                              void* d_out, int out_size, void* d_ws, size_t ws_size,
                              hipStream_t stream) {
    (void)in_sizes; (void)n_in; (void)out_size; (void)ws_size;

    const float*     x      = (const float*)d_in[0];
    const long long* eidx   = (const long long*)d_in[1];   // int64 [2,E]
    const float*     Wrel1  = (const float*)d_in[2];
    const float*     brel1  = (const float*)d_in[3];
    const float*     Wroot1 = (const float*)d_in[4];
    const float*     Wrel2  = (const float*)d_in[5];
    const float*     brel2  = (const float*)d_in[6];
    const float*     Wroot2 = (const float*)d_in[7];
    float*           out    = (float*)d_out;              // [N,1,64] flat

    const long long* src = eidx;            // row 0
    const long long* dst = eidx + N_EDGES;  // row 1

    // workspace: agg1 [N*128] | h [N*128];  y2 reuses agg1 region after gemm1
    float* agg1 = (float*)d_ws;
    float* hbuf = agg1 + (size_t)N_NODES * D_H;
    float* y2   = agg1;                     // [N,64], agg1 dead after gemm1

    // 1) agg1 = 0
    {
        int n = N_NODES * D_IN;
        gcn_zero_f32<<<(n + 255) / 256, 256, 0, stream>>>(agg1, n);
    }
    // 2) agg1[dst] += x[src]   (one wave per edge)
    {
        long long threads = (long long)N_EDGES * 32;
        int blocks = (int)((threads + 255) / 256);
        gcn_scatter128<<<blocks, 256, 0, stream>>>(x, src, dst, agg1);
    }
    // 3) h = relu(agg1@Wrel1^T + x@Wroot1^T + b1)
    gcn_gemm1<<<N_NODES / 16, 256, 0, stream>>>(agg1, x, Wrel1, Wroot1, brel1, hbuf);
    // 4) y2 = h@Wrel2^T ; out = h@Wroot2^T + b2
    gcn_gemm2<<<N_NODES / 16, 256, 0, stream>>>(hbuf, Wrel2, Wroot2, brel2, y2, out);
    // 5) out[dst] += y2[src]
    {
        long long threads = (long long)N_EDGES * 32;
        int blocks = (int)((threads + 255) / 256);
        gcn_scatter64<<<blocks, 256, 0, stream>>>(y2, src, dst, out);
    }
}